// TinyRGATLayer_30614526885986
// MI455X (gfx1250) — compile-verified
//
#include <hip/hip_runtime.h>
#include <hip/hip_bf16.h>
#include <math.h>

#define N_NODES 50000
#define N_EDGES 600000
#define HID 128
#define NUM_RELS 64
#define REL_DIM 16
#define LEAKY_SLOPE 0.2f
#define CONF_LOG_W 0.5f
#define WSTRIDE 132   // 132 mod 64 = 4 -> conflict-free b64 reads of W tiles

typedef float v2f __attribute__((ext_vector_type(2)));
typedef float v8f __attribute__((ext_vector_type(8)));

// ---------------------------------------------------------------------------
// H = x @ W_msg^T  (50000x128 @ 128x128), f32 WMMA 16x16x4.
// W_msg staged in LDS once per block (padded stride -> no bank conflicts).
// One wave computes a 16-row M slab across all 8 N-tiles; per K-step the A
// fragment is loaded once and 8 B fragments are batched into registers so
// the 8 WMMAs issue back-to-back instead of load->wait0->wmma serialization.
// ---------------------------------------------------------------------------
__global__ void gemm_H_kernel(const float* __restrict__ x,
                              const float* __restrict__ W,
                              float* __restrict__ H) {
    __shared__ float wlds[HID * WSTRIDE];          // 67,584 B (<320KB/WGP)

    // cooperative staged load of W_msg (coalesced global, padded LDS rows)
    for (int i = threadIdx.x; i < HID * HID; i += blockDim.x) {
        const int r = i >> 7;
        const int c = i & (HID - 1);
        wlds[r * WSTRIDE + c] = W[i];
    }
    __syncthreads();

    const int gwave = (blockIdx.x * blockDim.x + threadIdx.x) >> 5;
    const int lane  = threadIdx.x & 31;
    const int m0 = gwave * 16;
    if (m0 >= N_NODES) return;            // wave-uniform; after the barrier

    const int half = lane >> 4;           // 0: lanes 0-15, 1: lanes 16-31
    const int l16  = lane & 15;

    v8f zero = {0.f, 0.f, 0.f, 0.f, 0.f, 0.f, 0.f, 0.f};
    v8f acc[8];
#pragma unroll
    for (int t = 0; t < 8; ++t) acc[t] = zero;

    const float* arow = x + (size_t)(m0 + l16) * HID + 2 * half;

    for (int k0 = 0; k0 < HID; k0 += 4) {
        // A: 16x4 f32. lane l16 holds row m0+l16; K = k0 + 2*half + {0,1}
        v2f a = { arow[k0], arow[k0 + 1] };

        // batch all 8 B fragments (ds_load_b64, conflict-free) before WMMAs
        v2f b[8];
#pragma unroll
        for (int t = 0; t < 8; ++t) {
            const float* bp = &wlds[(t * 16 + l16) * WSTRIDE + k0 + 2 * half];
            b[t] = *(const v2f*)bp;       // dword offset is even -> 8B aligned
        }
#pragma unroll
        for (int t = 0; t < 8; ++t) {
            acc[t] = __builtin_amdgcn_wmma_f32_16x16x4_f32(
                false, a, false, b[t], (short)0, acc[t], false, false);
        }
    }

    // C/D layout: VGPR j, lanes 0-15 -> (M=j, N=l16); lanes 16-31 -> (M=j+8, N=l16)
#pragma unroll
    for (int t = 0; t < 8; ++t) {
        const int n0 = t * 16;
#pragma unroll
        for (int j = 0; j < 8; ++j) {
            const int m = m0 + j + half * 8;
            H[(size_t)m * HID + n0 + l16] = acc[t][j];
        }
    }
}

// ---------------------------------------------------------------------------
// Per-node attention scalars: sd[n] = H[n].a_dst, ss[n] = H[n].a_src
// ---------------------------------------------------------------------------
__global__ void node_scores_kernel(const float* __restrict__ H,
                                   const float* __restrict__ att,
                                   float* __restrict__ sd,
                                   float* __restrict__ ss) {
    const int node = (blockIdx.x * blockDim.x + threadIdx.x) >> 5;
    const int lane = threadIdx.x & 31;
    if (node >= N_NODES) return;
    const float* h = H + (size_t)node * HID;
    float ad = 0.f, as = 0.f;
#pragma unroll
    for (int i = 0; i < 4; ++i) {
        const int c = lane + 32 * i;
        const float v = h[c];
        ad += v * att[c];
        as += v * att[HID + c];
    }
#pragma unroll
    for (int off = 16; off > 0; off >>= 1) {
        ad += __shfl_xor(ad, off);
        as += __shfl_xor(as, off);
    }
    if (lane == 0) { sd[node] = ad; ss[node] = as; }
}

// ---------------------------------------------------------------------------
// Per-relation scalar: s_rel[k] = rel_emb[k] . (W_rel^T @ a_rel)   (64 values)
// ---------------------------------------------------------------------------
__global__ void rel_scores_kernel(const float* __restrict__ rel_emb,
                                  const float* __restrict__ W_rel,
                                  const float* __restrict__ att,
                                  float* __restrict__ s_rel) {
    __shared__ float v[REL_DIM];
    const int t = threadIdx.x;
    if (t < REL_DIM) {
        float s = 0.f;
        for (int h = 0; h < HID; ++h) s += W_rel[h * REL_DIM + t] * att[2 * HID + h];
        v[t] = s;
    }
    __syncthreads();
    if (t < NUM_RELS) {
        float s = 0.f;
#pragma unroll
        for (int d = 0; d < REL_DIM; ++d) s += rel_emb[t * REL_DIM + d] * v[d];
        s_rel[t] = s;
    }
}

// ---------------------------------------------------------------------------
// Init: out accumulator = 0, e_max = -inf, denom = 0
// ---------------------------------------------------------------------------
__global__ void init_kernel(float* __restrict__ out,
                            float* __restrict__ e_max,
                            float* __restrict__ denom) {
    const int i = blockIdx.x * blockDim.x + threadIdx.x;
    if (i < N_NODES) { e_max[i] = -__builtin_inff(); denom[i] = 0.f; }
    if (i < N_NODES * HID) out[i] = 0.f;
}

__device__ __forceinline__ void atomicMaxFloat(float* addr, float value) {
    if (value >= 0.f)
        atomicMax((int*)addr, __float_as_int(value));
    else
        atomicMin((unsigned int*)addr, __float_as_uint(value));
}

// ---------------------------------------------------------------------------
// Edge pass A: e_raw + segment max (thread per edge)
// ---------------------------------------------------------------------------
__global__ void edge_logits_kernel(const int* __restrict__ src,
                                   const int* __restrict__ dst,
                                   const int* __restrict__ etype,
                                   const float* __restrict__ conf,
                                   const float* __restrict__ sd,
                                   const float* __restrict__ ss,
                                   const float* __restrict__ s_rel,
                                   float* __restrict__ e_raw,
                                   float* __restrict__ e_max) {
    const int e = blockIdx.x * blockDim.x + threadIdx.x;
    if (e >= N_EDGES) return;
    const int s = src[e];
    const int d = dst[e];
    int t = etype[e];
    t = t < 0 ? 0 : (t > NUM_RELS - 1 ? NUM_RELS - 1 : t);
    float er = sd[d] + ss[s] + s_rel[t];
    er = er >= 0.f ? er : LEAKY_SLOPE * er;
    float c = conf[e];
    c = c < 1e-6f ? 1e-6f : c;
    er += CONF_LOG_W * logf(c);
    e_raw[e] = er;
    atomicMaxFloat(&e_max[d], er);
}

// ---------------------------------------------------------------------------
// Edge pass B: ex = exp(e_raw - max); denom[dst] += ex; out[dst] += ex*H[src]
// One wave per edge; 128-float row handled as 4 coalesced atomics per lane.
// ---------------------------------------------------------------------------
__global__ void edge_scatter_kernel(const int* __restrict__ src,
                                    const int* __restrict__ dst,
                                    const float* __restrict__ e_raw,
                                    const float* __restrict__ e_max,
                                    const float* __restrict__ H,
                                    float* __restrict__ denom,
                                    float* __restrict__ out) {
    const int e    = (blockIdx.x * blockDim.x + threadIdx.x) >> 5;
    const int lane = threadIdx.x & 31;
    if (e >= N_EDGES) return;
    const int s = src[e];
    const int d = dst[e];
    float ex = 0.f;
    if (lane == 0) {
        ex = expf(e_raw[e] - e_max[d]);
        atomicAdd(&denom[d], ex);
    }
    ex = __shfl(ex, 0);
    const float* hs = H + (size_t)s * HID;
    float* od = out + (size_t)d * HID;
#pragma unroll
    for (int i = 0; i < 4; ++i) {
        const int c = lane + 32 * i;
        atomicAdd(&od[c], ex * hs[c]);
    }
}

// ---------------------------------------------------------------------------
// Finalize: out = out/denom + bias  (denom==0 -> bias, matches empty segment)
// ---------------------------------------------------------------------------
__global__ void finalize_kernel(const float* __restrict__ denom,
                                const float* __restrict__ bias,
                                float* __restrict__ out) {
    const int i = blockIdx.x * blockDim.x + threadIdx.x;
    if (i >= N_NODES * HID) return;
    const int n = i >> 7;
    const int h = i & (HID - 1);
    const float dn = denom[n];
    const float v = dn > 0.f ? out[i] / dn : 0.f;
    out[i] = v + bias[h];
}

static inline size_t alignUp(size_t v, size_t a) { return (v + a - 1) & ~(a - 1); }

extern "C" void kernel_launch(void* const* d_in, const int* in_sizes, int n_in,
                              void* d_out, int out_size, void* d_ws, size_t ws_size,
                              hipStream_t stream) {
    const float* x       = (const float*)d_in[0];
    const float* W_msg   = (const float*)d_in[1];
    const float* rel_emb = (const float*)d_in[2];
    const float* W_rel   = (const float*)d_in[3];
    const float* att_vec = (const float*)d_in[4];
    const float* bias    = (const float*)d_in[5];
    const float* e_attr  = (const float*)d_in[6];
    const int*   e_index = (const int*)d_in[7];     // (2, E): row 0 = src, row 1 = dst
    const int*   e_type  = (const int*)d_in[8];
    float*       out     = (float*)d_out;

    const int* src = e_index;
    const int* dst = e_index + N_EDGES;

    // Workspace carve-up (~28.8 MB total)
    char* w = (char*)d_ws;
    float* H     = (float*)w; w += alignUp((size_t)N_NODES * HID * sizeof(float), 256);
    float* sd    = (float*)w; w += alignUp((size_t)N_NODES * sizeof(float), 256);
    float* ss    = (float*)w; w += alignUp((size_t)N_NODES * sizeof(float), 256);
    float* s_rel = (float*)w; w += alignUp((size_t)NUM_RELS * sizeof(float), 256);
    float* e_raw = (float*)w; w += alignUp((size_t)N_EDGES * sizeof(float), 256);
    float* e_max = (float*)w; w += alignUp((size_t)N_NODES * sizeof(float), 256);
    float* denom = (float*)w; w += alignUp((size_t)N_NODES * sizeof(float), 256);

    const int BLK = 256;

    // 1) H = x @ W_msg^T : 3125 M-tiles, one wave each (8 waves/block)
    {
        const int waves  = (N_NODES + 15) / 16;          // 3125
        const int blocks = (waves * 32 + BLK - 1) / BLK; // 391
        gemm_H_kernel<<<blocks, BLK, 0, stream>>>(x, W_msg, H);
    }
    // 2) per-node score scalars (wave per node)
    {
        const int blocks = ((size_t)N_NODES * 32 + BLK - 1) / BLK;
        node_scores_kernel<<<blocks, BLK, 0, stream>>>(H, att_vec, sd, ss);
    }
    // 3) per-relation scalars
    rel_scores_kernel<<<1, 64, 0, stream>>>(rel_emb, W_rel, att_vec, s_rel);
    // 4) init accumulators
    {
        const int blocks = (N_NODES * HID + BLK - 1) / BLK;
        init_kernel<<<blocks, BLK, 0, stream>>>(out, e_max, denom);
    }
    // 5) edge logits + segment max
    {
        const int blocks = (N_EDGES + BLK - 1) / BLK;
        edge_logits_kernel<<<blocks, BLK, 0, stream>>>(src, dst, e_type, e_attr,
                                                       sd, ss, s_rel, e_raw, e_max);
    }
    // 6) exp + denom + weighted scatter of H[src] (wave per edge)
    {
        const int blocks = ((size_t)N_EDGES * 32 + BLK - 1) / BLK;
        edge_scatter_kernel<<<blocks, BLK, 0, stream>>>(src, dst, e_raw, e_max,
                                                        H, denom, out);
    }
    // 7) normalize + bias
    {
        const int blocks = (N_NODES * HID + BLK - 1) / BLK;
        finalize_kernel<<<blocks, BLK, 0, stream>>>(denom, bias, out);
    }
    (void)in_sizes; (void)n_in; (void)out_size; (void)ws_size;
}